// Dyanmicatt_1786706395355
// MI455X (gfx1250) — compile-verified
//
#include <hip/hip_runtime.h>
#include <math.h>

typedef __attribute__((ext_vector_type(2))) float v2f;
typedef __attribute__((ext_vector_type(8))) float v8f;

namespace {
constexpr int kD = 512;                      // feature dim
constexpr int kT = 128;                      // timesteps
constexpr int kB = 2048;                     // batch
constexpr int kRows = kB * kT;               // 262144 rows of x / gi
constexpr int kLdsStride = kD + 4;           // 516 floats -> conflict-free DS banks
constexpr int kWavesPerBlock = 4;            // 128-thread blocks
constexpr int kRowsPerWave = 32;             // two 16-row WMMA tiles per wave
constexpr int kNWaves2 = kB / 32;            // 64 scan waves
}

// ---------------------------------------------------------------------------
// Kernel 1: gi[r][g] = sum_k x[r][k] * w_ih[g][k]   (r = b*T + t, g = 0..2)
// via V_WMMA_F32_16X16X4_F32, B padded to 16 columns (cols 3..15 = 0).
// ---------------------------------------------------------------------------
__global__ __launch_bounds__(128)
void gi_gemm_wmma(const float* __restrict__ x, const float* __restrict__ w_ih,
                  float* __restrict__ gi) {
  __shared__ float wl[16 * kLdsStride];
  for (int i = threadIdx.x; i < 16 * kLdsStride; i += blockDim.x) {
    const int row = i / kLdsStride;
    const int k   = i - row * kLdsStride;
    wl[i] = (row < 3 && k < kD) ? w_ih[row * kD + k] : 0.0f;
  }
  __syncthreads();

  const int lane = threadIdx.x & 31;
  const int wave = threadIdx.x >> 5;
  const int col  = lane & 15;     // A: row-in-tile;  B: gate column (only 0..2 live)
  const int kh   = lane >> 4;     // K sub-offset: lanes 16..31 hold K+2, K+3

  const long row0 =
      ((long)blockIdx.x * kWavesPerBlock + wave) * (long)kRowsPerWave;

  const float* a0p = x + (row0 + col) * kD + 2 * kh;         // tile 0 (rows 0..15)
  const float* a1p = x + (row0 + 16 + col) * kD + 2 * kh;    // tile 1 (rows 16..31)
  const float* bp  = wl + col * kLdsStride + 2 * kh;         // shared B operand

  v8f c0 = {};
  v8f c1 = {};
  #pragma unroll 4
  for (int kk = 0; kk < kD; kk += 4) {
    v2f a0 = __builtin_nontemporal_load((const v2f*)(a0p + kk));
    v2f a1 = __builtin_nontemporal_load((const v2f*)(a1p + kk));
    v2f b  = *(const v2f*)(bp + kk);
    c0 = __builtin_amdgcn_wmma_f32_16x16x4_f32(false, a0, false, b,
                                               (short)0, c0, false, false);
    c1 = __builtin_amdgcn_wmma_f32_16x16x4_f32(false, a1, false, b,
                                               (short)0, c1, false, false);
  }

  // C/D layout: VGPR r holds row r (lanes 0-15) and row r+8 (lanes 16-31),
  // N = lane&15. Only gate columns 0..2 are live.
  if (col < 3) {
    const long r0 = row0 + 8 * kh;
    const long r1 = row0 + 16 + 8 * kh;
    #pragma unroll
    for (int r = 0; r < 8; ++r) {
      gi[(r0 + r) * 3 + col] = c0[r];
      gi[(r1 + r) * 3 + col] = c1[r];
    }
  }
}

// ---------------------------------------------------------------------------
// Kernel 2: per-batch GRU scan (hidden size 1) + per-wave batch reduction.
// partial[waveId][t] = sum over that wave's 32 batches of h_t.
// ---------------------------------------------------------------------------
__global__ __launch_bounds__(256)
void gru_scan(const float* __restrict__ gi, const float* __restrict__ h0,
              const float* __restrict__ w_hh, const float* __restrict__ b_ih,
              const float* __restrict__ b_hh, float* __restrict__ partial) {
  const int b      = blockIdx.x * blockDim.x + threadIdx.x;   // batch index
  const int lane   = threadIdx.x & 31;
  const int waveId = b >> 5;

  const float wr = w_hh[0], wz = w_hh[1], wn = w_hh[2];
  const float br = b_hh[0], bz = b_hh[1], bn = b_hh[2];
  const float ar = b_ih[0], az = b_ih[1], an = b_ih[2];

  float h = h0[b];
  const float* g = gi + (long)b * kT * 3;

  for (int t = 0; t < kT; ++t) {
    const float gir = g[t * 3 + 0] + ar;
    const float giz = g[t * 3 + 1] + az;
    const float gin = g[t * 3 + 2] + an;
    const float ghr = fmaf(h, wr, br);
    const float ghz = fmaf(h, wz, bz);
    const float ghn = fmaf(h, wn, bn);
    const float r = 1.0f / (1.0f + expf(-(gir + ghr)));
    const float z = 1.0f / (1.0f + expf(-(giz + ghz)));
    const float n = tanhf(gin + r * ghn);
    h = (1.0f - z) * n + z * h;

    float s = h;
    #pragma unroll
    for (int off = 16; off > 0; off >>= 1)
      s += __shfl_xor(s, off, 32);
    if (lane == 0) partial[waveId * kT + t] = s;
  }
}

// ---------------------------------------------------------------------------
// Kernel 3: out[t] = (1/B) * sum over 64 wave partials.
// ---------------------------------------------------------------------------
__global__ __launch_bounds__(128)
void finalize(const float* __restrict__ partial, float* __restrict__ out) {
  const int t = threadIdx.x;
  if (t < kT) {
    float s = 0.0f;
    #pragma unroll 8
    for (int w = 0; w < kNWaves2; ++w) s += partial[w * kT + t];
    out[t] = s * (1.0f / (float)kB);
  }
}

extern "C" void kernel_launch(void* const* d_in, const int* in_sizes, int n_in,
                              void* d_out, int out_size, void* d_ws, size_t ws_size,
                              hipStream_t stream) {
  (void)in_sizes; (void)n_in; (void)out_size; (void)ws_size;
  const float* x    = (const float*)d_in[0];
  const float* h0   = (const float*)d_in[1];
  const float* w_ih = (const float*)d_in[2];
  const float* w_hh = (const float*)d_in[3];
  const float* b_ih = (const float*)d_in[4];
  const float* b_hh = (const float*)d_in[5];
  float* out = (float*)d_out;

  float* gi      = (float*)d_ws;                       // kRows * 3 floats (3 MB)
  float* partial = gi + (long)kRows * 3;               // 64 * 128 floats

  const int blocks1 = kRows / (kWavesPerBlock * kRowsPerWave);  // 2048
  gi_gemm_wmma<<<blocks1, 128, 0, stream>>>(x, w_ih, gi);

  gru_scan<<<kB / 256, 256, 0, stream>>>(gi, h0, w_hh, b_ih, b_hh, partial);

  finalize<<<1, 128, 0, stream>>>(partial, out);
}